// VisionEncoderMambaBlock_48816598286474
// MI455X (gfx1250) — compile-verified
//
#include <hip/hip_runtime.h>
#include <hip/hip_bf16.h>
#include <math.h>

// ---------------------------------------------------------------------------
// VisionEncoderMambaBlock for gfx1250 (MI455X), wave32 + WMMA + TDM.
// B=4, S=2048, D=512, DT_RANK=32, N=16.  fwd-GEMM + first SSM are dead code.
// ---------------------------------------------------------------------------

#define Bq 4
#define Sq 2048
#define Dq 512
#define DTR 32
#define Nst 16
#define ROWS (Bq * Sq)          // 8192

typedef float v2f __attribute__((ext_vector_type(2)));
typedef float v8f __attribute__((ext_vector_type(8)));
typedef unsigned int u32x4 __attribute__((ext_vector_type(4)));
typedef int i32x4 __attribute__((ext_vector_type(4)));
typedef int i32x8 __attribute__((ext_vector_type(8)));

// ----------------------------- LayerNorm -----------------------------------
__global__ void ln_kernel(const float* __restrict__ x,
                          const float* __restrict__ gamma,
                          const float* __restrict__ beta,
                          float* __restrict__ xn)
{
    __shared__ float wsum[8], wsq[8];
    __shared__ float s_mu, s_rs;
    const int row = blockIdx.x;                 // 0..8191
    const int tid = threadIdx.x;                // 256 threads, 2 elems each
    const float* xr = x + (size_t)row * Dq;

    float v0 = xr[tid];
    float v1 = xr[tid + 256];
    float s  = v0 + v1;
    float s2 = v0 * v0 + v1 * v1;
    #pragma unroll
    for (int m = 1; m < 32; m <<= 1) {
        s  += __shfl_xor(s,  m, 32);
        s2 += __shfl_xor(s2, m, 32);
    }
    if ((tid & 31) == 0) { wsum[tid >> 5] = s; wsq[tid >> 5] = s2; }
    __syncthreads();
    if (tid == 0) {
        float t = 0.f, t2 = 0.f;
        #pragma unroll
        for (int i = 0; i < 8; ++i) { t += wsum[i]; t2 += wsq[i]; }
        float mu  = t * (1.0f / Dq);
        float var = t2 * (1.0f / Dq) - mu * mu;
        s_mu = mu;
        s_rs = rsqrtf(var + 1e-5f);
    }
    __syncthreads();
    const float mu = s_mu, rs = s_rs;
    float* xo = xn + (size_t)row * Dq;
    xo[tid]       = (v0 - mu) * rs * gamma[tid]       + beta[tid];
    xo[tid + 256] = (v1 - mu) * rs * gamma[tid + 256] + beta[tid + 256];
}

// ------------------------ Generic WMMA f32 GEMM -----------------------------
// C[M x N] = act(A[M x K(lda)] @ W[K x N] + bias).  act: 0=none, 1=softplus.
// Block = 256 threads = 8 waves; each wave owns a 16x16 tile; block covers
// 128 rows x 16 cols.  The K x 16 W panel is DMA'd into LDS by the Tensor
// Data Mover (one descriptor, issued by wave 0, tracked with TENSORcnt),
// then shared by all 8 waves.
__global__ void gemm_wmma_f32(const float* __restrict__ A, int lda,
                              const float* __restrict__ W,
                              const float* __restrict__ bias,
                              float* __restrict__ C,
                              int N, int K, int act)
{
    extern __shared__ float Wt[];               // K x 16 floats (tile row-major)
    const int tid  = threadIdx.x;
    const int wave = tid >> 5;
    const int lane = tid & 31;
    const int col0 = blockIdx.y * 16;
    const int row0 = blockIdx.x * 128 + wave * 16;

    // --- TDM: load the 2D tile W[0:K, col0:col0+16] into LDS -----------------
    if (wave == 0) {                            // uniform branch: one wave issues
        const unsigned long long ga = (unsigned long long)(W + col0); // tile start
        u32x4 g0;
        g0[0] = 1u;                                             // count=1, user mode
        g0[1] = 0u;                                             // lds_addr (dyn-LDS base)
        g0[2] = (unsigned int)(ga & 0xffffffffull);             // global_addr[31:0]
        g0[3] = (unsigned int)((ga >> 32) & 0x01fffffful)       // global_addr[56:32]
              | (2u << 30);                                     // type = 2 ("image")
        i32x8 g1;
        g1[0] = 2 << 16;                                        // data_size = 4 bytes
        g1[1] = (N & 0xffff) << 16;                             // tensor_dim0 lo16
        g1[2] = ((unsigned)N >> 16) | ((K & 0xffff) << 16);     // dim0 hi | dim1 lo
        g1[3] = ((unsigned)K >> 16) | (16 << 16);               // dim1 hi | tile_dim0=16
        g1[4] = K;                                              // tile_dim1=K, tile_dim2=0
        g1[5] = N;                                              // dim0_stride lo32
        g1[6] = 0;                                              // dim0_stride hi | dim1_stride lo
        g1[7] = 0;                                              // dim1_stride hi
        i32x4 gz4 = {0, 0, 0, 0};
        i32x8 gz8 = {0, 0, 0, 0, 0, 0, 0, 0};
        __builtin_amdgcn_tensor_load_to_lds(g0, g1, gz4, gz4, gz8, 0);
        __builtin_amdgcn_s_wait_tensorcnt(0);
    }
    __syncthreads();

    const int half = lane >> 4;                 // 0 or 1
    const int l16  = lane & 15;
    const int koff = half * 2;                  // K sub-offset per ISA layout
    const float* arow = A + (size_t)(row0 + l16) * lda;

    v8f c = {0.f, 0.f, 0.f, 0.f, 0.f, 0.f, 0.f, 0.f};
    #pragma unroll 4
    for (int kk = 0; kk < K; kk += 4) {
        v2f a, b;
        const float* ap = arow + kk + koff;     // 8B-aligned pair
        a.x = ap[0];
        a.y = ap[1];
        b.x = Wt[(kk + koff) * 16 + l16];
        b.y = Wt[(kk + koff + 1) * 16 + l16];
        c = __builtin_amdgcn_wmma_f32_16x16x4_f32(
                false, a, false, b, (short)0, c, false, false);
    }

    const float bv = bias ? bias[col0 + l16] : 0.f;
    #pragma unroll
    for (int v = 0; v < 8; ++v) {
        const int row = row0 + v + half * 8;
        float val = c[v] + bv;
        if (act == 1)                            // softplus
            val = (val > 20.f) ? val : log1pf(__expf(val));
        C[(size_t)row * N + col0 + l16] = val;
    }
}

// -------------------- Selective scan + silu combine -------------------------
// One lane per (b, d, n); 16 consecutive lanes share (b, d).
// h_s = exp(delta*A)*h_{s-1} + (delta*u)*Bm ;  y = sum_n h*Cm + D*u
// out = silu(z1) * (z1 + y) + x
__global__ void scan_fused_kernel(const float* __restrict__ delta_buf,
                                  const float* __restrict__ u_buf,
                                  const float* __restrict__ dbc,
                                  const float* __restrict__ z1,
                                  const float* __restrict__ x,
                                  const float* __restrict__ A_log,
                                  const float* __restrict__ D_ssm,
                                  float* __restrict__ out)
{
    const int t = blockIdx.x * blockDim.x + threadIdx.x;   // 0..32767
    const int n = t & 15;
    const int d = (t >> 4) & (Dq - 1);
    const int b = t >> 13;

    const float a   = -__expf(A_log[d * Nst + n]);
    const float dsm = D_ssm[d];
    float h = 0.f;
    const size_t rowb = (size_t)b * Sq;

    for (int s = 0; s < Sq; ++s) {
        const size_t r = rowb + s;
        const float delta = delta_buf[r * Dq + d];
        const float u     = u_buf[r * Dq + d];
        const float Bm    = dbc[r * 64 + DTR + n];
        const float Cm    = dbc[r * 64 + DTR + Nst + n];

        const float dA = __expf(delta * a);
        h = fmaf(dA, h, delta * u * Bm);

        float p = h * Cm;                       // reduce over n (16 lanes)
        p += __shfl_xor(p, 1, 32);
        p += __shfl_xor(p, 2, 32);
        p += __shfl_xor(p, 4, 32);
        p += __shfl_xor(p, 8, 32);

        if (n == 0) {
            const float y   = p + dsm * u;
            const float zv  = z1[r * Dq + d];
            const float sil = zv / (1.f + __expf(-zv));
            out[r * Dq + d] = sil * (zv + y) + x[r * Dq + d];
        }
    }
}

// ---------------------------------------------------------------------------
extern "C" void kernel_launch(void* const* d_in, const int* in_sizes, int n_in,
                              void* d_out, int out_size, void* d_ws, size_t ws_size,
                              hipStream_t stream) {
    (void)in_sizes; (void)n_in; (void)out_size; (void)ws_size;

    const float* x      = (const float*)d_in[0];
    const float* gamma  = (const float*)d_in[1];
    const float* beta   = (const float*)d_in[2];
    const float* W_proj = (const float*)d_in[3];
    const float* b_proj = (const float*)d_in[4];
    // d_in[5], d_in[6] = W_fwd, b_fwd : dead code in the reference
    const float* W_bwd  = (const float*)d_in[7];
    const float* b_bwd  = (const float*)d_in[8];
    const float* W_dbc  = (const float*)d_in[9];
    const float* W_dt   = (const float*)d_in[10];
    const float* b_dt   = (const float*)d_in[11];
    const float* A_log  = (const float*)d_in[12];
    const float* D_ssm  = (const float*)d_in[13];
    float* out = (float*)d_out;

    const size_t BSD = (size_t)ROWS * Dq;       // 4.19M floats
    float* xn   = (float*)d_ws;                 // slot A (reused as delta)
    float* z1   = xn + BSD;                     // slot B
    float* u    = z1 + BSD;                     // slot C
    float* dbc  = u + BSD;                      // 8192 x 64
    float* delta = xn;                          // xn dead after z1 GEMM

    // 1) LayerNorm
    ln_kernel<<<ROWS, 256, 0, stream>>>(x, gamma, beta, xn);

    // 2) z1 = xn @ W_proj + b_proj
    gemm_wmma_f32<<<dim3(ROWS / 128, Dq / 16), 256, Dq * 16 * sizeof(float), stream>>>(
        xn, Dq, W_proj, b_proj, z1, Dq, Dq, 0);

    // 3) u = z1 @ W_bwd + b_bwd
    gemm_wmma_f32<<<dim3(ROWS / 128, Dq / 16), 256, Dq * 16 * sizeof(float), stream>>>(
        z1, Dq, W_bwd, b_bwd, u, Dq, Dq, 0);

    // 4) dbc = u @ W_dbc            (8192 x 512 @ 512 x 64, no bias)
    gemm_wmma_f32<<<dim3(ROWS / 128, 64 / 16), 256, Dq * 16 * sizeof(float), stream>>>(
        u, Dq, W_dbc, nullptr, dbc, 64, Dq, 0);

    // 5) delta = softplus(dbc[:, :32] @ W_dt + b_dt)
    gemm_wmma_f32<<<dim3(ROWS / 128, Dq / 16), 256, DTR * 16 * sizeof(float), stream>>>(
        dbc, 64, W_dt, b_dt, delta, Dq, DTR, 1);

    // 6) selective scan + silu gate + residual, fused
    scan_fused_kernel<<<(Bq * Dq * Nst) / 256, 256, 0, stream>>>(
        delta, u, dbc, z1, x, A_log, D_ssm, out);
}